// Four_Pos_Fusion_Embedding_85495618995109
// MI455X (gfx1250) — compile-verified
//
#include <hip/hip_runtime.h>

typedef float v2f __attribute__((ext_vector_type(2)));
typedef float v4f __attribute__((ext_vector_type(4)));
typedef float v8f __attribute__((ext_vector_type(8)));

#define B_    8
#define C_    256
#define W_    96
#define H_    128
#define TAB_  1025
#define FILL_ 510

// ---------------------------------------------------------------------------
// Kernel 1: P[tab] = pe_tab (1025x128) @ W_f[:, tab*128:(tab+1)*128]^T (128x128)
// fp32-exact via V_WMMA_F32_16X16X4_F32. One wave per 16x16 output tile.
// ---------------------------------------------------------------------------
__global__ void __launch_bounds__(256)
precompute_tables(const float* __restrict__ pe_ss,
                  const float* __restrict__ pe_se,
                  const float* __restrict__ pe_es,
                  const float* __restrict__ pe_ee,
                  const float* __restrict__ Wf,   // (H_, 4*H_) row-major
                  float* __restrict__ P) {        // 4 tables of (TAB_, H_)
  const int lane = threadIdx.x & 31;
  const int wave = threadIdx.x >> 5;
  const int gid  = blockIdx.x * 8 + wave;

  const int NT_TILES  = (TAB_ + 15) / 16;      // 65
  const int NH_TILES  = H_ / 16;               // 8
  const int PER_TABLE = NT_TILES * NH_TILES;   // 520
  if (gid >= 4 * PER_TABLE) return;            // wave-uniform: EXEC stays all-1s

  const int table = gid / PER_TABLE;
  const int rem   = gid % PER_TABLE;
  const int t0    = (rem / NH_TILES) * 16;
  const int h0    = (rem % NH_TILES) * 16;

  const float* pe = (table == 0) ? pe_ss
                  : (table == 1) ? pe_se
                  : (table == 2) ? pe_es : pe_ee;
  const int coff = table * H_;                 // column block of W_f

  // A-frag (16x4 f32): lane L, vgpr v -> M = L%16, K = v + 2*(L/16)
  int arow = t0 + (lane & 15);
  if (arow > TAB_ - 1) arow = TAB_ - 1;        // clamp pad rows (not stored)
  const int   klo  = (lane >> 4) * 2;
  const float* aptr = pe + (size_t)arow * H_ + klo;
  // B-frag (4x16 f32): lane L, vgpr v -> N = L%16, K = v + 2*(L/16)
  // B[k,n] = W_f[h0+n, coff+k]  (row-major W_f => contiguous float2 in k)
  const float* bptr = Wf + (size_t)(h0 + (lane & 15)) * (4 * H_) + coff + klo;

  v8f acc = {0.f, 0.f, 0.f, 0.f, 0.f, 0.f, 0.f, 0.f};
  #pragma unroll 8
  for (int k = 0; k < H_; k += 4) {
    v2f a = *(const v2f*)(aptr + k);
    v2f b = *(const v2f*)(bptr + k);
    acc = __builtin_amdgcn_wmma_f32_16x16x4_f32(
        /*neg_a=*/false, a, /*neg_b=*/false, b,
        /*c_mod=*/(short)0, acc, /*reuse_a=*/false, /*reuse_b=*/false);
  }

  // D layout: lane L, vgpr v -> N = L%16, M = v + 8*(L/16)
  float* Pt = P + (size_t)table * TAB_ * H_;
  #pragma unroll
  for (int v = 0; v < 8; ++v) {
    const int trow = t0 + v + 8 * (lane >> 4);
    if (trow < TAB_) Pt[(size_t)trow * H_ + h0 + (lane & 15)] = acc[v];
  }
}

// ---------------------------------------------------------------------------
// Kernel 2: per (b,c,w) row -> 4 gathers into precomputed tables, sum, +bias,
// ReLU, non-temporal streamed store. One wave per row, float4 per lane.
// Tables (2.1 MB) stay L2-resident; output (~100 MB) streams out NT.
// ---------------------------------------------------------------------------
__global__ void __launch_bounds__(256)
gather_fuse(const int* __restrict__ pos_s, const int* __restrict__ pos_e,
            const int* __restrict__ lex_s, const int* __restrict__ lex_e,
            const int* __restrict__ seq_len, const int* __restrict__ lex_num,
            const float* __restrict__ P, const float* __restrict__ b_f,
            float* __restrict__ out) {
  const int lane = threadIdx.x & 31;
  const int wave = threadIdx.x >> 5;
  const int row  = blockIdx.x * 8 + wave;      // < B_*C_*W_ = 196608
  const int b = row / (C_ * W_);
  const int r = row % (C_ * W_);
  const int i = r / W_;
  const int j = r % W_;

  const int ps = pos_s[b * C_ + i];
  const int pq = pos_e[b * C_ + i];
  const int ls = lex_s[b * W_ + j];
  const int le = lex_e[b * W_ + j];
  const bool m = (i < seq_len[b]) && (j < lex_num[b]);

  int dss = m ? (ps - ls) : FILL_;
  int dee = m ? (pq - le) : FILL_;
  if (j == 0) { dss = 0; dee = 0; }            // .at[:,:,0].set(0) after mask
  int dse = ps - le;
  int des = pq - ls;
  if (dss < 0) dss += TAB_;                    // negative-index wrap
  if (dse < 0) dse += TAB_;
  if (des < 0) des += TAB_;
  if (dee < 0) dee += TAB_;

  const int h = lane * 4;
  const float* Pss = P;
  const float* Pse = P + 1 * TAB_ * H_;
  const float* Pes = P + 2 * TAB_ * H_;
  const float* Pee = P + 3 * TAB_ * H_;

  v4f s = *(const v4f*)(Pss + (size_t)dss * H_ + h);
  s = s + *(const v4f*)(Pse + (size_t)dse * H_ + h);
  s = s + *(const v4f*)(Pes + (size_t)des * H_ + h);
  s = s + *(const v4f*)(Pee + (size_t)dee * H_ + h);
  s = s + *(const v4f*)(b_f + h);
  #pragma unroll
  for (int q = 0; q < 4; ++q) s[q] = s[q] > 0.f ? s[q] : 0.f;

  __builtin_nontemporal_store(s, (v4f*)(out + (size_t)row * H_ + h));
}

// ---------------------------------------------------------------------------
extern "C" void kernel_launch(void* const* d_in, const int* in_sizes, int n_in,
                              void* d_out, int out_size, void* d_ws, size_t ws_size,
                              hipStream_t stream) {
  const int*   pos_s   = (const int*)  d_in[0];
  const int*   pos_e   = (const int*)  d_in[1];
  const int*   lex_s   = (const int*)  d_in[2];
  const int*   lex_e   = (const int*)  d_in[3];
  const int*   seq_len = (const int*)  d_in[4];
  const int*   lex_num = (const int*)  d_in[5];
  const float* pe_ss   = (const float*)d_in[6];
  const float* pe_se   = (const float*)d_in[7];
  const float* pe_es   = (const float*)d_in[8];
  const float* pe_ee   = (const float*)d_in[9];
  const float* W_f     = (const float*)d_in[10];
  const float* b_f     = (const float*)d_in[11];
  float*       out     = (float*)d_out;
  float*       P       = (float*)d_ws;   // 4 * 1025 * 128 floats = 2.1 MB

  // Pass 1: fold W_f into the four tables (WMMA fp32). 2080 waves total.
  precompute_tables<<<260, 256, 0, stream>>>(pe_ss, pe_se, pe_es, pe_ee, W_f, P);

  // Pass 2: bandwidth-bound gather + add + ReLU. 196608 rows, 8 rows/block.
  const int R = B_ * C_ * W_;            // 196608
  gather_fuse<<<R / 8, 256, 0, stream>>>(pos_s, pos_e, lex_s, lex_e,
                                         seq_len, lex_num, P, b_f, out);
}